// SSM_model_10557029614109
// MI455X (gfx1250) — compile-verified
//
#include <hip/hip_runtime.h>
#include <hip/hip_bf16.h>

#define LSEQ   16384
#define NPOLE  64
#define NBATCH 1024
#define STEPSZ 0.01
#define CHUNK  64
#define NCHUNK (NBATCH / CHUNK)
#define PI_D   3.14159265358979323846
#define TWO_PI_F 6.28318530717958647692f

typedef __attribute__((ext_vector_type(2))) float v2f;
typedef __attribute__((ext_vector_type(8))) float v8f;
typedef unsigned int u32;
typedef __attribute__((ext_vector_type(4))) u32 su4;
typedef __attribute__((ext_vector_type(8))) u32 su8;

struct cdbl { double x, y; };
__device__ __forceinline__ cdbl cadd(cdbl a, cdbl b){ cdbl r; r.x=a.x+b.x; r.y=a.y+b.y; return r; }
__device__ __forceinline__ cdbl csub(cdbl a, cdbl b){ cdbl r; r.x=a.x-b.x; r.y=a.y-b.y; return r; }
__device__ __forceinline__ cdbl cmul(cdbl a, cdbl b){ cdbl r; r.x=a.x*b.x-a.y*b.y; r.y=a.x*b.y+a.y*b.x; return r; }
__device__ __forceinline__ cdbl cdiv(cdbl a, cdbl b){
  double d = b.x*b.x + b.y*b.y;
  cdbl r; r.x=(a.x*b.x+a.y*b.y)/d; r.y=(a.y*b.x-a.x*b.y)/d; return r;
}
__device__ __forceinline__ cdbl cscale(double s, cdbl a){ cdbl r; r.x=s*a.x; r.y=s*a.y; return r; }

// ---- Tensor Data Mover: DMA one 2D tile (global, strided) -> LDS (padded) ----
// lds_addr : workgroup-relative LDS byte address (low 32 bits of generic ptr)
// tile_w   : tile width in 4-byte elements, tile_h: rows
// stride   : row stride of the source matrix in elements
// pad_code : LDS pad interval code (pads 1 DWORD every 2^(code+1) DWORDs)
__device__ __forceinline__ void tdm_load_tile(u32 lds_addr, const float* gsrc,
                                              u32 tile_w, u32 tile_h,
                                              u32 stride, u32 pad_code) {
  unsigned long long ga = (unsigned long long)(uintptr_t)gsrc;
  su4 g0;
  g0[0] = 1u;                                   // count=1, user descriptor
  g0[1] = lds_addr;                             // lds_addr [63:32]
  g0[2] = (u32)ga;                              // global_addr lo32
  g0[3] = (u32)((ga >> 32) & 0x1FFFFFFu) | 0x80000000u; // global_addr[56:32] | type=2
  su8 g1;
  const u32 td0 = 1u << 20;                     // tensor dims huge: no OOB clipping
  const u32 td1 = 1u << 20;
  g1[0] = (2u << 16) | (1u << 20) | (pad_code << 22); // data_size=4B, pad_en, pad_interval, pad_amount=1 DW
  g1[1] = (td0 & 0xFFFFu) << 16;                // atomic_barrier_addr=0 | tensor_dim0 lo16
  g1[2] = ((td0 >> 16) & 0xFFFFu) | ((td1 & 0xFFFFu) << 16); // tensor_dim0 hi | tensor_dim1 lo
  g1[3] = ((td1 >> 16) & 0xFFFFu) | (tile_w << 16);          // tensor_dim1 hi | tile_dim0
  g1[4] = tile_h;                               // tile_dim1 | tile_dim2=0
  g1[5] = stride;                               // tensor_dim0_stride lo32 (elements)
  g1[6] = 0u;                                   // stride hi16 | tensor_dim1_stride lo16
  g1[7] = 0u;
  asm volatile("tensor_load_to_lds %0, %1" :: "s"(g0), "s"(g1) : "memory");
}

// ---- Stage 0: coefficient vectors r00..r11 = f(C(I-A_L), B, P, Q) in fp64 ----
__global__ void ssm_k0_coeffs(const cdbl* __restrict__ A_L, const cdbl* __restrict__ Bv,
                              const cdbl* __restrict__ Cv, const cdbl* __restrict__ Pv,
                              const cdbl* __restrict__ Qv, cdbl* __restrict__ coef) {
  int n = threadIdx.x;
  if (n >= NPOLE) return;
  cdbl acc; acc.x = 0.0; acc.y = 0.0;
  for (int m = 0; m < NPOLE; ++m) acc = cadd(acc, cmul(Cv[m], A_L[m*NPOLE + n]));
  cdbl c_ = csub(Cv[n], acc);
  cdbl qc; qc.x = Qv[n].x; qc.y = -Qv[n].y;
  coef[0*NPOLE + n] = cmul(c_, Bv[n]);
  coef[1*NPOLE + n] = cmul(c_, Pv[n]);
  coef[2*NPOLE + n] = cmul(qc, Bv[n]);
  coef[3*NPOLE + n] = cmul(qc, Pv[n]);
}

// ---- Stage 1: K_w spectrum bins 0..L/2 (Cauchy sums, fp64) ----
__global__ void ssm_k1_kw(const cdbl* __restrict__ coef, const cdbl* __restrict__ diag,
                          float2* __restrict__ Kw) {
  int j = blockIdx.x * blockDim.x + threadIdx.x;
  if (j > LSEQ/2) return;
  double th = (2.0 * PI_D) * (double)j / (double)LSEQ;
  cdbl z;   z.x = cos(th);        z.y = -sin(th);
  cdbl opz; opz.x = 1.0 + z.x;    opz.y =  z.y;
  cdbl omz; omz.x = 1.0 - z.x;    omz.y = -z.y;
  cdbl w = cscale(2.0 / STEPSZ, cdiv(omz, opz));
  cdbl s00, s01, s10, s11;
  s00.x=s00.y=s01.x=s01.y=s10.x=s10.y=s11.x=s11.y=0.0;
  for (int n = 0; n < NPOLE; ++n) {
    cdbl d = csub(w, diag[n]);
    double den = d.x*d.x + d.y*d.y;
    cdbl inv; inv.x = d.x/den; inv.y = -d.y/den;
    s00 = cadd(s00, cmul(coef[0*NPOLE+n], inv));
    s01 = cadd(s01, cmul(coef[1*NPOLE+n], inv));
    s10 = cadd(s10, cmul(coef[2*NPOLE+n], inv));
    s11 = cadd(s11, cmul(coef[3*NPOLE+n], inv));
  }
  cdbl onep; onep.x = 1.0 + s11.x; onep.y = s11.y;
  cdbl kv  = csub(s00, cdiv(cmul(s01, s10), onep));
  cdbl Kwv = cdiv(cscale(2.0, kv), opz);
  Kw[j] = make_float2((float)Kwv.x, (float)Kwv.y);
}

// ---- Stage 2: K = irfft(K_w, L); integer-reduced angles (exact mod 2^14) ----
__global__ void ssm_k2_irfft(const float2* __restrict__ Kw, float* __restrict__ Kt) {
  int n = blockIdx.x * blockDim.x + threadIdx.x;
  if (n >= LSEQ) return;
  float acc = Kw[0].x + ((n & 1) ? -Kw[LSEQ/2].x : Kw[LSEQ/2].x);
  for (int j = 1; j < LSEQ/2; ++j) {
    int idx = (j * n) & (LSEQ - 1);
    float th = (TWO_PI_F / (float)LSEQ) * (float)idx;
    float s, c; __sincosf(th, &s, &c);
    acc += 2.0f * (Kw[j].x * c - Kw[j].y * s);
  }
  Kt[n] = acc / (float)LSEQ;
}

// ---- Stage 3: Khat2L[k] = rfft(k_pad, 2L) bins 0..L/2 ----
__global__ void ssm_k3_khat(const float* __restrict__ Kt, float2* __restrict__ Khat) {
  int k = blockIdx.x * blockDim.x + threadIdx.x;
  if (k > LSEQ/2) return;
  float ar = 0.f, ai = 0.f;
  for (int n = 0; n < LSEQ; ++n) {
    int idx = (k * n) & (2*LSEQ - 1);
    float th = (TWO_PI_F / (float)(2*LSEQ)) * (float)idx;
    float s, c; __sincosf(th, &s, &c);
    float kv = Kt[n];
    ar += kv * c;
    ai -= kv * s;
  }
  Khat[k] = make_float2(ar, ai);
}

// ---- Stage 4: fold irfft weights a_m into Khat, reshape to [k1(64) x k2(256)] ----
__global__ void ssm_k4_khatc(const float2* __restrict__ Khat, float2* __restrict__ KhatC) {
  int t = blockIdx.x * blockDim.x + threadIdx.x;
  if (t >= 64*256) return;
  int k1 = t >> 8, k2 = t & 255;
  int m = k2 + 256*k1;
  float2 v = make_float2(0.f, 0.f);
  if (m <= LSEQ/2) {
    float a = (m == 0 || m == LSEQ/2) ? (1.0f/(float)LSEQ) : (2.0f/(float)LSEQ);
    float2 kh = Khat[m];
    v = make_float2(a*kh.x, a*kh.y);
  }
  KhatC[t] = v;
}

// ---- Stage 5: constant DFT / twiddle factor tables for the four-step DFTs ----
__global__ void ssm_k5_tables(float* __restrict__ Ere, float* __restrict__ Eim,
                              float2* __restrict__ TWF,
                              float* __restrict__ DtRe, float* __restrict__ DtIm, float* __restrict__ DtImN,
                              float* __restrict__ GRe,  float* __restrict__ GIm,  float* __restrict__ GImN,
                              float2* __restrict__ TWI,
                              float* __restrict__ Hre,  float* __restrict__ Himn) {
  int t = blockIdx.x * blockDim.x + threadIdx.x;
  if (t >= 32768) return;
  { // E[n2][k2] = exp(-2pi i n2 k2 / 256)
    int n2 = t >> 8, k2 = t & 255;
    float th = (TWO_PI_F/256.f) * (float)((n2*k2) & 255);
    float s, c; __sincosf(th, &s, &c);
    Ere[t] = c; Eim[t] = -s;
  }
  { // TWF[n1][k2] = exp(-2pi i n1 k2 / 32768)
    int n1 = t >> 8, k2 = t & 255;
    float th = (TWO_PI_F/32768.f) * (float)((n1*k2) & 32767);
    float s, c; __sincosf(th, &s, &c);
    TWF[t] = make_float2(c, -s);
  }
  { // TWI[k2][n1] = exp(+2pi i k2 n1 / 16384)
    int k2 = t >> 7, n1 = t & 127;
    float th = (TWO_PI_F/16384.f) * (float)((k2*n1) & 16383);
    float s, c; __sincosf(th, &s, &c);
    TWI[t] = make_float2(c, s);
  }
  { // H[k2][n2]: exp(+2pi i k2 n2 / 128); real part and negated imag
    int k2 = t >> 7, n2 = t & 127;
    float th = (TWO_PI_F/128.f) * (float)((k2*n2) & 127);
    float s, c; __sincosf(th, &s, &c);
    Hre[t] = c; Himn[t] = -s;
  }
  if (t < 8192) { // Dt[n1][k1] = exp(-2pi i k1 n1 / 128), k1 > 32 zero-padded
    int n1 = t >> 6, k1 = t & 63;
    float re = 0.f, im = 0.f, imn = 0.f;
    if (k1 <= 32) {
      float th = (TWO_PI_F/128.f) * (float)((k1*n1) & 127);
      float s, c; __sincosf(th, &s, &c);
      re = c; im = -s; imn = s;
    }
    DtRe[t] = re; DtIm[t] = im; DtImN[t] = imn;
  }
  if (t < 8192) { // G[k1][n1] = exp(+2pi i k1 n1 / 64), k1 > 32 zero-padded
    int k1 = t >> 7, n1 = t & 127;
    float re = 0.f, im = 0.f, imn = 0.f;
    if (k1 <= 32) {
      float th = (TWO_PI_F/64.f) * (float)((k1*n1) & 63);
      float s, c; __sincosf(th, &s, &c);
      re = c; im = s; imn = -s;
    }
    GRe[t] = re; GIm[t] = im; GImN[t] = imn;
  }
}

// ---- Gather x rows into [ (r,n1) x n2 ] layout for four-step stage F1 ----
__global__ void ssm_k6_gather(const float* __restrict__ x, float* __restrict__ Xg, int b0) {
  int t = blockIdx.x * blockDim.x + threadIdx.x;
  if (t >= CHUNK * LSEQ) return;
  int r = t >> 14, n = t & (LSEQ - 1);
  int n1 = n & 127, n2 = n >> 7;
  Xg[(r*128 + n1)*128 + n2] = x[(size_t)(b0 + r)*LSEQ + n];
}

// ---- Twiddle + transpose (forward): T[(r,n1),k2]*TWF -> T2[(r,k2),n1] ----
__global__ void ssm_k7_twtrF(const float* __restrict__ Tre, const float* __restrict__ Tim,
                             const float2* __restrict__ TWF,
                             float* __restrict__ T2re, float* __restrict__ T2im) {
  int t = blockIdx.x * blockDim.x + threadIdx.x;
  if (t >= CHUNK*128*256) return;
  int row = t >> 8, k2 = t & 255;
  int r = row >> 7, n1 = row & 127;
  float tr = Tre[t], ti = Tim[t];
  float2 w = TWF[n1*256 + k2];
  int o = (r*256 + k2)*128 + n1;
  T2re[o] = tr*w.x - ti*w.y;
  T2im[o] = tr*w.y + ti*w.x;
}

// ---- Pointwise spectrum product Yc = Xhat * KhatC ----
__global__ void ssm_k8_point(const float* __restrict__ XhRe, const float* __restrict__ XhIm,
                             const float2* __restrict__ KhatC,
                             float* __restrict__ YcRe, float* __restrict__ YcIm) {
  int t = blockIdx.x * blockDim.x + threadIdx.x;
  if (t >= CHUNK*256*64) return;
  int row = t >> 6, k1 = t & 63;
  int k2 = row & 255;
  float2 kc = KhatC[k1*256 + k2];
  float xr = XhRe[t], xi = XhIm[t];
  YcRe[t] = xr*kc.x - xi*kc.y;
  YcIm[t] = xr*kc.y + xi*kc.x;
}

// ---- Twiddle + transpose (inverse): U[(r,k2),n1]*TWI -> U2[(r,n1),k2] ----
__global__ void ssm_k9_twtrI(const float* __restrict__ Ure, const float* __restrict__ Uim,
                             const float2* __restrict__ TWI,
                             float* __restrict__ U2re, float* __restrict__ U2im) {
  int t = blockIdx.x * blockDim.x + threadIdx.x;
  if (t >= CHUNK*256*128) return;
  int row = t >> 7, n1 = t & 127;
  int r = row >> 8, k2 = row & 255;
  float ur = Ure[t], ui = Uim[t];
  float2 w = TWI[k2*128 + n1];
  int o = (r*128 + n1)*256 + k2;
  U2re[o] = ur*w.x - ui*w.y;
  U2im[o] = ur*w.y + ui*w.x;
}

// ---- Final: tanh(h + D*x), scatter back to row-major output ----
__global__ void ssm_k10_tanh(const float* __restrict__ yout, const float* __restrict__ x,
                             const float* __restrict__ Dv, float* __restrict__ out, int b0) {
  int t = blockIdx.x * blockDim.x + threadIdx.x;
  if (t >= CHUNK * LSEQ) return;
  int row = t >> 7, n2 = t & 127;
  int r = row >> 7, n1 = row & 127;
  int n = n1 + 128*n2;
  size_t gi = (size_t)(b0 + r)*LSEQ + n;
  out[gi] = tanhf(yout[t] + Dv[0]*x[gi]);
}

// ---- Core: f32 WMMA GEMM; A/B tiles DMAed to LDS by the Tensor Data Mover ----
__launch_bounds__(128)
__global__ void ssm_gemm_wmma(const float* __restrict__ A, const float* __restrict__ B,
                              float* __restrict__ C, int K, int lda, int ldb, int ldc,
                              int accumulate) {
  __shared__ float As[64][17];   // pad 1 DWORD / 16 -> TDM pad_interval code 3
  __shared__ float Bs[16][65];   // pad 1 DWORD / 64 -> TDM pad_interval code 5
  const int tid  = threadIdx.x;
  const int wave = tid >> 5;
  const int lane = tid & 31;
  const int half = lane >> 4;
  const int l    = lane & 15;
  const int m_block = blockIdx.y * 64;
  const int n_block = blockIdx.x * 64;
  const u32 ldsA = (u32)(uintptr_t)&As[0][0];   // low 32 bits = LDS byte address
  const u32 ldsB = (u32)(uintptr_t)&Bs[0][0];

  v8f acc[4];
  if (accumulate) {
    #pragma unroll
    for (int j = 0; j < 4; ++j)
      #pragma unroll
      for (int r = 0; r < 8; ++r)
        acc[j][r] = C[(size_t)(m_block + wave*16 + r + 8*half)*ldc + n_block + j*16 + l];
  } else {
    #pragma unroll
    for (int j = 0; j < 4; ++j)
      #pragma unroll
      for (int r = 0; r < 8; ++r)
        acc[j][r] = 0.0f;
  }

  for (int k0 = 0; k0 < K; k0 += 16) {
    if (k0 + 16 < K)                     // warm L2 for the next A tile strip
      __builtin_prefetch(A + (size_t)(m_block + (tid & 63))*lda + k0 + 16, 0, 0);
    if (tid < 32) {
      // wave 0 drives the TDM (issues once per wave, EXEC-independent)
      tdm_load_tile(ldsA, A + (size_t)m_block*lda + k0, 16u, 64u, (u32)lda, 3u);
      tdm_load_tile(ldsB, B + (size_t)k0*ldb + n_block, 64u, 16u, (u32)ldb, 5u);
      __builtin_amdgcn_s_wait_tensorcnt(0);
    }
    __syncthreads();
    #pragma unroll
    for (int kb = 0; kb < 16; kb += 4) {
      v2f a;  // A 16x4 fragment: lanes0-15 K={kb,kb+1}, lanes16-31 K={kb+2,kb+3}
      a.x = As[wave*16 + l][kb + 2*half];
      a.y = As[wave*16 + l][kb + 2*half + 1];
      #pragma unroll
      for (int j = 0; j < 4; ++j) {
        v2f b; // B 4x16 fragment: same K striping, N = column lane
        b.x = Bs[kb + 2*half][j*16 + l];
        b.y = Bs[kb + 2*half + 1][j*16 + l];
        acc[j] = __builtin_amdgcn_wmma_f32_16x16x4_f32(
            false, a, false, b, (short)0, acc[j], false, false);
      }
    }
    __syncthreads();
  }

  #pragma unroll
  for (int j = 0; j < 4; ++j)
    #pragma unroll
    for (int r = 0; r < 8; ++r)
      C[(size_t)(m_block + wave*16 + r + 8*half)*ldc + n_block + j*16 + l] = acc[j][r];
}

extern "C" void kernel_launch(void* const* d_in, const int* in_sizes, int n_in,
                              void* d_out, int out_size, void* d_ws, size_t ws_size,
                              hipStream_t stream) {
  (void)in_sizes; (void)n_in; (void)out_size; (void)ws_size;
  const float* x   = (const float*)d_in[0];
  const cdbl* A_L  = (const cdbl*)d_in[1];
  const cdbl* Bv   = (const cdbl*)d_in[2];
  const cdbl* Cv   = (const cdbl*)d_in[3];
  const cdbl* Pv   = (const cdbl*)d_in[4];
  const cdbl* Qv   = (const cdbl*)d_in[5];
  const cdbl* diag = (const cdbl*)d_in[6];
  const float* Dv  = (const float*)d_in[7];
  float* out = (float*)d_out;

  char* w = (char*)d_ws;
  size_t off = 0;
  auto alloc = [&](size_t bytes) -> void* {
    void* p = (void*)(w + off);
    off += (bytes + 255) & ~(size_t)255;
    return p;
  };
  cdbl*   coef  = (cdbl*)  alloc(4*NPOLE*sizeof(cdbl));
  float2* Kw    = (float2*)alloc((LSEQ/2+1)*sizeof(float2));
  float*  Kt    = (float*) alloc(LSEQ*sizeof(float));
  float2* Khat  = (float2*)alloc((LSEQ/2+1)*sizeof(float2));
  float2* KhatC = (float2*)alloc(64*256*sizeof(float2));
  float*  Ere   = (float*) alloc(32768*sizeof(float));
  float*  Eim   = (float*) alloc(32768*sizeof(float));
  float2* TWF   = (float2*)alloc(32768*sizeof(float2));
  float*  DtRe  = (float*) alloc(8192*sizeof(float));
  float*  DtIm  = (float*) alloc(8192*sizeof(float));
  float*  DtImN = (float*) alloc(8192*sizeof(float));
  float*  GRe   = (float*) alloc(8192*sizeof(float));
  float*  GIm   = (float*) alloc(8192*sizeof(float));
  float*  GImN  = (float*) alloc(8192*sizeof(float));
  float2* TWI   = (float2*)alloc(32768*sizeof(float2));
  float*  Hre   = (float*) alloc(32768*sizeof(float));
  float*  Himn  = (float*) alloc(32768*sizeof(float));
  float*  BA    = (float*) alloc((size_t)2*1048576*sizeof(float)); // Xg / Yc
  float*  BB    = (float*) alloc((size_t)2*2097152*sizeof(float)); // T  / U
  float*  BC    = (float*) alloc((size_t)2*2097152*sizeof(float)); // T2 / U2
  float*  BD    = (float*) alloc((size_t)2*1048576*sizeof(float)); // Xh / yout

  ssm_k0_coeffs<<<dim3(1), dim3(64), 0, stream>>>(A_L, Bv, Cv, Pv, Qv, coef);
  ssm_k1_kw    <<<dim3(33), dim3(256), 0, stream>>>(coef, diag, Kw);
  ssm_k2_irfft <<<dim3(64), dim3(256), 0, stream>>>(Kw, Kt);
  ssm_k3_khat  <<<dim3(33), dim3(256), 0, stream>>>(Kt, Khat);
  ssm_k4_khatc <<<dim3(64), dim3(256), 0, stream>>>(Khat, KhatC);
  ssm_k5_tables<<<dim3(128), dim3(256), 0, stream>>>(Ere, Eim, TWF, DtRe, DtIm, DtImN,
                                                     GRe, GIm, GImN, TWI, Hre, Himn);

  float* Xg   = BA;
  float* YcRe = BA;            float* YcIm = BA + 1048576;
  float* Tre  = BB;            float* Tim  = BB + 2097152;
  float* Ure  = BB;            float* Uim  = BB + 2097152;
  float* T2re = BC;            float* T2im = BC + 2097152;
  float* U2re = BC;            float* U2im = BC + 2097152;
  float* XhRe = BD;            float* XhIm = BD + 1048576;
  float* yout = BD;

  for (int c = 0; c < NCHUNK; ++c) {
    int b0 = c * CHUNK;
    ssm_k6_gather<<<dim3(CHUNK*LSEQ/256), dim3(256), 0, stream>>>(x, Xg, b0);

    // F1: T = Xg[8192x128] * E[128x256]  (real * complex -> 2 GEMMs)
    dim3 gF1(256/64, (CHUNK*128)/64);
    ssm_gemm_wmma<<<gF1, dim3(128), 0, stream>>>(Xg, Ere, Tre, 128, 128, 256, 256, 0);
    ssm_gemm_wmma<<<gF1, dim3(128), 0, stream>>>(Xg, Eim, Tim, 128, 128, 256, 256, 0);

    ssm_k7_twtrF<<<dim3(CHUNK*128*256/256), dim3(256), 0, stream>>>(Tre, Tim, TWF, T2re, T2im);

    // F3: Xh = T2[16384x128] * Dt[128x64]  (complex * complex -> 4 GEMMs)
    dim3 gF3(64/64, (CHUNK*256)/64);
    ssm_gemm_wmma<<<gF3, dim3(128), 0, stream>>>(T2re, DtRe,  XhRe, 128, 128, 64, 64, 0);
    ssm_gemm_wmma<<<gF3, dim3(128), 0, stream>>>(T2im, DtImN, XhRe, 128, 128, 64, 64, 1);
    ssm_gemm_wmma<<<gF3, dim3(128), 0, stream>>>(T2re, DtIm,  XhIm, 128, 128, 64, 64, 0);
    ssm_gemm_wmma<<<gF3, dim3(128), 0, stream>>>(T2im, DtRe,  XhIm, 128, 128, 64, 64, 1);

    ssm_k8_point<<<dim3(CHUNK*256*64/256), dim3(256), 0, stream>>>(XhRe, XhIm, KhatC, YcRe, YcIm);

    // I1: U = Yc[16384x64] * G[64x128]  (complex * complex -> 4 GEMMs)
    dim3 gI1(128/64, (CHUNK*256)/64);
    ssm_gemm_wmma<<<gI1, dim3(128), 0, stream>>>(YcRe, GRe,  Ure, 64, 64, 128, 128, 0);
    ssm_gemm_wmma<<<gI1, dim3(128), 0, stream>>>(YcIm, GImN, Ure, 64, 64, 128, 128, 1);
    ssm_gemm_wmma<<<gI1, dim3(128), 0, stream>>>(YcRe, GIm,  Uim, 64, 64, 128, 128, 0);
    ssm_gemm_wmma<<<gI1, dim3(128), 0, stream>>>(YcIm, GRe,  Uim, 64, 64, 128, 128, 1);

    ssm_k9_twtrI<<<dim3(CHUNK*256*128/256), dim3(256), 0, stream>>>(Ure, Uim, TWI, U2re, U2im);

    // I3: yout = Re( U2[8192x256] * H[256x128] )  (2 GEMMs)
    dim3 gI3(128/64, (CHUNK*128)/64);
    ssm_gemm_wmma<<<gI3, dim3(128), 0, stream>>>(U2re, Hre,  yout, 256, 256, 128, 128, 0);
    ssm_gemm_wmma<<<gI3, dim3(128), 0, stream>>>(U2im, Himn, yout, 256, 256, 128, 128, 1);

    ssm_k10_tanh<<<dim3(CHUNK*LSEQ/256), dim3(256), 0, stream>>>(yout, x, Dv, out, b0);
  }
}